// TransformerBlock_20435454395094
// MI455X (gfx1250) — compile-verified
//
#include <hip/hip_runtime.h>
#include <math.h>

// ---------------------------------------------------------------------------
// MI455X (gfx1250) implementation of the TransformerBlock reference.
// Bandwidth-bound workload (~12 GFLOP vs ~1GB traffic @ 23.3 TB/s): we keep
// full fp32 precision and run all channel-mixing GEMMs on the fp32 matrix
// path V_WMMA_F32_16X16X4_F32 (16x16 out tile per wave, K stepped by 4).
// ---------------------------------------------------------------------------

typedef __attribute__((ext_vector_type(2))) float v2f;
typedef __attribute__((ext_vector_type(8))) float v8f;

static constexpr int B_    = 2;
static constexpr int DIM_  = 48;
static constexpr int H_    = 256;
static constexpr int W_    = 256;
static constexpr int HW_   = H_ * W_;
static constexpr int HEADS_ = 8;
static constexpr int CHEAD_ = 6;
static constexpr int GSPLIT = 32;

// ---- workspace layout (floats) --------------------------------------------
static constexpr size_t SZ48   = (size_t)B_ * DIM_ * HW_;        // 6,291,456
static constexpr size_t SZ144  = (size_t)B_ * 144 * HW_;         // 18,874,368
static constexpr size_t SZ288  = (size_t)B_ * 288 * HW_;         // 37,748,736
static constexpr size_t OFF_X2  = 0;                  // x after attention residual
static constexpr size_t OFF_XN  = OFF_X2 + SZ48;      // xn (LN1) then xn2 (LN2)
static constexpr size_t OFF_OC  = OFF_XN + SZ48;      // out_conv
static constexpr size_t OFF_AO  = OFF_OC + SZ48;      // attention output
static constexpr size_t OFF_Q   = OFF_AO + SZ48;      // qkv post-depthwise; later yg[ B,96,HW ]
static constexpr size_t OFF_BIG = OFF_Q + SZ144;      // qkv_pre -> f_conv -> y (pin out)
static constexpr size_t OFF_T1  = OFF_BIG + SZ288;    // conv branch stage outs
static constexpr size_t OFF_T2  = OFF_T1 + (size_t)B_ * DIM_ * 64 * 64;
static constexpr size_t OFF_T3  = OFF_T2 + (size_t)B_ * DIM_ * 16 * 16;
static constexpr size_t OFF_XD  = OFF_T3 + (size_t)B_ * DIM_ * 4 * 4;  // 96
static constexpr size_t OFF_Q0  = OFF_XD + 96;                         // 96
static constexpr size_t OFF_NRM = OFF_Q0 + 96;                         // 192
static constexpr size_t OFF_AW  = OFF_NRM + 192;                       // 576 attn weights
static constexpr size_t OFF_AP  = OFF_AW + 576;                        // GSPLIT*16*36 partial grams

__device__ __forceinline__ float gelu_exact(float x) {
    return 0.5f * x * (1.0f + erff(x * 0.70710678118654752440f));
}

// ---------------------------------------------------------------------------
// Generic channel GEMM via fp32 WMMA:  Y[b,m,p] = sum_k W[m,k]*X[b,k,p]
// (+bias, +res1, +res2).  One wave = one 16x16 (m x p) tile; K stepped by 4.
// grid: (HW/64, M/16, B), block: 128 (4 waves, each a different p-tile).
// ---------------------------------------------------------------------------
__global__ void wmma_mm_k(const float* __restrict__ Wm, const float* __restrict__ X,
                          float* __restrict__ Y, const float* __restrict__ bias,
                          const float* __restrict__ res1, const float* __restrict__ res2,
                          int M, int K)
{
    const int wave = threadIdx.x >> 5;
    const int lane = threadIdx.x & 31;
    const int half = lane >> 4;        // 0: lanes 0-15, 1: lanes 16-31
    const int l16  = lane & 15;
    const int n0 = (blockIdx.x * (blockDim.x >> 5) + wave) * 16;
    const int m0 = blockIdx.y * 16;
    const size_t xbase = (size_t)blockIdx.z * K * HW_;
    const size_t ybase = (size_t)blockIdx.z * M * HW_;

    v8f acc = {};
    const float* wrow = Wm + (size_t)(m0 + l16) * K;   // A: M=l16(+0/), K pairs
    for (int k0 = 0; k0 < K; k0 += 4) {
        const int ka = k0 + half * 2;
        v2f a, bm;
        a.x = wrow[ka];                                // A[M][ka], A[M][ka+1]
        a.y = wrow[ka + 1];
        const float* xp = X + xbase + (size_t)ka * HW_ + n0 + l16;
        bm.x = xp[0];                                  // B[ka][n], B[ka+1][n]
        bm.y = xp[HW_];
        acc = __builtin_amdgcn_wmma_f32_16x16x4_f32(
            false, a, false, bm, (short)0, acc, false, false);
    }
#pragma unroll
    for (int r = 0; r < 8; ++r) {
        const int m = m0 + r + half * 8;               // C/D: rows r / r+8
        const size_t idx = ybase + (size_t)m * HW_ + n0 + l16;
        float v = acc[r];
        if (bias) v += bias[m];
        if (res1) v += res1[idx];
        if (res2) v += res2[idx];
        Y[idx] = v;
    }
}

// ---------------------------------------------------------------------------
// LayerNorm over channels (48) per spatial position.
// ---------------------------------------------------------------------------
__global__ void ln_k(const float* __restrict__ x, const float* __restrict__ w,
                     const float* __restrict__ b, float* __restrict__ y)
{
    const size_t gid = (size_t)blockIdx.x * blockDim.x + threadIdx.x;
    if (gid >= (size_t)B_ * HW_) return;
    const int bb = (int)(gid / HW_);
    const int p  = (int)(gid % HW_);
    const float* xp = x + (size_t)bb * DIM_ * HW_ + p;
    float v[DIM_];
    float mu = 0.f;
#pragma unroll
    for (int c = 0; c < DIM_; ++c) { v[c] = xp[(size_t)c * HW_]; mu += v[c]; }
    mu *= (1.0f / DIM_);
    float var = 0.f;
#pragma unroll
    for (int c = 0; c < DIM_; ++c) { const float d = v[c] - mu; var += d * d; }
    var *= (1.0f / DIM_);
    const float rstd = rsqrtf(var + 1e-5f);
    float* yp = y + (size_t)bb * DIM_ * HW_ + p;
#pragma unroll
    for (int c = 0; c < DIM_; ++c)
        yp[(size_t)c * HW_] = (v[c] - mu) * rstd * w[c] + b[c];
}

// ---------------------------------------------------------------------------
// Depthwise 3x3 conv, pad = dil, for C channels on the 256x256 grid.
// ---------------------------------------------------------------------------
__global__ void dwconv3_k(const float* __restrict__ in, const float* __restrict__ w,
                          float* __restrict__ out, int C, int dil)
{
    const size_t gid = (size_t)blockIdx.x * blockDim.x + threadIdx.x;
    const size_t total = (size_t)B_ * C * HW_;
    if (gid >= total) return;
    const int p = (int)(gid % HW_);
    const int c = (int)((gid / HW_) % C);
    const int bb = (int)(gid / ((size_t)HW_ * C));
    const int x = p & (W_ - 1), y = p >> 8;
    const float* ip = in + ((size_t)bb * C + c) * HW_;
    const float* wp = w + (size_t)c * 9;
    float s = 0.f;
#pragma unroll
    for (int ky = 0; ky < 3; ++ky)
#pragma unroll
        for (int kx = 0; kx < 3; ++kx) {
            const int yy = y + (ky - 1) * dil, xx = x + (kx - 1) * dil;
            if (yy >= 0 && yy < H_ && xx >= 0 && xx < W_)
                s += wp[ky * 3 + kx] * ip[yy * W_ + xx];
        }
    out[gid] = s;
}

// ---------------------------------------------------------------------------
// fc mixing over the raw-memory reshape:  f_conv[b, c*9+o, p] =
//   fc_b[o] + sum_j fc_w[o,j] * qkv_flat[b, p*144 + j*6 + c]
// A tile of 64 positions (64*144 contiguous floats) is staged in LDS.
// ---------------------------------------------------------------------------
__global__ void fc_k(const float* __restrict__ qkv, const float* __restrict__ fcw,
                     const float* __restrict__ fcb, float* __restrict__ fconv)
{
    __shared__ float tile[64 * 144];
    const int bb = blockIdx.y;
    const int p0 = blockIdx.x * 64;
    const float* src = qkv + (size_t)bb * 144 * HW_ + (size_t)p0 * 144;
    for (int i = threadIdx.x; i < 64 * 144; i += blockDim.x) tile[i] = src[i];
    __syncthreads();
    const int pl = threadIdx.x;          // blockDim == 64
    const int p2 = p0 + pl;
    const float* row = &tile[pl * 144];
    for (int c = 0; c < 6; ++c)
        for (int o = 0; o < 9; ++o) {
            float s = fcb[o];
#pragma unroll
            for (int j = 0; j < 24; ++j) s += fcw[o * 24 + j] * row[j * 6 + c];
            fconv[((size_t)bb * 54 + c * 9 + o) * HW_ + p2] = s;
        }
}

// ---------------------------------------------------------------------------
// Grouped conv (groups = chead = 6): out[b,o,p], in-group 9 channels, 3x3 pad 1.
// ---------------------------------------------------------------------------
__global__ void depconv_k(const float* __restrict__ fconv, const float* __restrict__ w,
                          const float* __restrict__ bias, float* __restrict__ out)
{
    const size_t gid = (size_t)blockIdx.x * blockDim.x + threadIdx.x;
    if (gid >= (size_t)B_ * DIM_ * HW_) return;
    const int p = (int)(gid % HW_);
    const int o = (int)((gid / HW_) % DIM_);
    const int bb = (int)(gid / ((size_t)HW_ * DIM_));
    const int g = o >> 3;                 // 48 out / 6 groups = 8 per group
    const int x = p & (W_ - 1), y = p >> 8;
    const float* wp = w + (size_t)o * 81; // [o][i(9)][ky][kx]
    float s = bias[o];
    for (int i = 0; i < 9; ++i) {
        const float* ip = fconv + ((size_t)bb * 54 + g * 9 + i) * HW_;
#pragma unroll
        for (int ky = 0; ky < 3; ++ky)
#pragma unroll
            for (int kx = 0; kx < 3; ++kx) {
                const int yy = y + ky - 1, xx = x + kx - 1;
                if (yy >= 0 && yy < H_ && xx >= 0 && xx < W_)
                    s += wp[i * 9 + ky * 3 + kx] * ip[yy * W_ + xx];
            }
    }
    out[gid] = s;
}

// ---------------------------------------------------------------------------
// Row L2 norms for q (qkv ch 0..47) and k (qkv ch 48..95) over HW.
// ---------------------------------------------------------------------------
__global__ void rownorm_k(const float* __restrict__ qkv, float* __restrict__ norms)
{
    const int bb = blockIdx.x / 96, ch = blockIdx.x % 96;
    const float* row = qkv + ((size_t)bb * 144 + ch) * HW_;
    float s = 0.f;
    for (int p = threadIdx.x; p < HW_; p += blockDim.x) { const float v = row[p]; s += v * v; }
    __shared__ float red[256];
    red[threadIdx.x] = s;
    __syncthreads();
    for (int o = 128; o > 0; o >>= 1) {
        if (threadIdx.x < o) red[threadIdx.x] += red[threadIdx.x + o];
        __syncthreads();
    }
    if (threadIdx.x == 0) norms[blockIdx.x] = fmaxf(sqrtf(red[0]), 1e-12f);
}

// ---------------------------------------------------------------------------
// Split-K Gram: attnP[(split*16+bh)*36 + c*6+d] = partial sum q_c * k_d.
// Deterministic (no atomics); softmax kernel folds the splits.
// ---------------------------------------------------------------------------
__global__ void gram_k(const float* __restrict__ qkv, float* __restrict__ attnP)
{
    const int bh = blockIdx.x, split = blockIdx.y;
    const int bb = bh >> 3, h = bh & 7;
    const float* q  = qkv + ((size_t)bb * 144 + h * CHEAD_) * HW_;
    const float* kk = qkv + ((size_t)bb * 144 + 48 + h * CHEAD_) * HW_;
    float acc[36];
#pragma unroll
    for (int i = 0; i < 36; ++i) acc[i] = 0.f;
    const int stride = blockDim.x * gridDim.y;
    for (int p = split * blockDim.x + threadIdx.x; p < HW_; p += stride) {
        float qa[6], kb[6];
#pragma unroll
        for (int c = 0; c < 6; ++c) { qa[c] = q[(size_t)c * HW_ + p]; kb[c] = kk[(size_t)c * HW_ + p]; }
#pragma unroll
        for (int c = 0; c < 6; ++c)
#pragma unroll
            for (int d = 0; d < 6; ++d) acc[c * 6 + d] += qa[c] * kb[d];
    }
    __shared__ float red[256];
    for (int i = 0; i < 36; ++i) {
        red[threadIdx.x] = acc[i];
        __syncthreads();
        for (int o = 128; o > 0; o >>= 1) {
            if (threadIdx.x < o) red[threadIdx.x] += red[threadIdx.x + o];
            __syncthreads();
        }
        if (threadIdx.x == 0) attnP[((size_t)split * 16 + bh) * 36 + i] = red[0];
        __syncthreads();
    }
}

// ---------------------------------------------------------------------------
// Fold Gram partials, apply norms/temperature/q0 outer-product, softmax over 6.
// ---------------------------------------------------------------------------
__global__ void softmax_k(const float* __restrict__ attnP, const float* __restrict__ norms,
                          const float* __restrict__ temp, const float* __restrict__ q0,
                          float* __restrict__ attnW)
{
    const int t = threadIdx.x;
    if (t >= B_ * HEADS_ * CHEAD_) return;
    const int bb = t / (HEADS_ * CHEAD_);
    const int h  = (t / CHEAD_) % HEADS_;
    const int c  = t % CHEAD_;
    const int bh = bb * 8 + h;
    const float nq  = norms[bb * 96 + h * 6 + c];
    const float tm  = temp[h];
    const float q0c = q0[bb * 48 + h * 6 + c];
    float logit[6];
    float mx = -1e30f;
    for (int d = 0; d < 6; ++d) {
        float s = 0.f;
        for (int sp = 0; sp < GSPLIT; ++sp) s += attnP[((size_t)sp * 16 + bh) * 36 + c * 6 + d];
        const float nk = norms[bb * 96 + 48 + h * 6 + d];
        float a = s / (nq * nk) * tm;
        a *= q0c * q0[bb * 48 + h * 6 + d];
        logit[d] = a;
        mx = fmaxf(mx, a);
    }
    float se = 0.f;
    for (int d = 0; d < 6; ++d) { logit[d] = expf(logit[d] - mx); se += logit[d]; }
    const float inv = 1.0f / se;
    for (int d = 0; d < 6; ++d) attnW[(size_t)bh * 36 + c * 6 + d] = logit[d] * inv;
}

// ---------------------------------------------------------------------------
// out[b, h*6+c, p] = sum_d attnW[b,h,c,d] * v[b, 96+h*6+d, p]
// ---------------------------------------------------------------------------
__global__ void attnv_k(const float* __restrict__ qkv, const float* __restrict__ attnW,
                        float* __restrict__ ao)
{
    const size_t gid = (size_t)blockIdx.x * blockDim.x + threadIdx.x;
    if (gid >= (size_t)B_ * DIM_ * HW_) return;
    const int p = (int)(gid % HW_);
    const int ch = (int)((gid / HW_) % DIM_);
    const int bb = (int)(gid / ((size_t)HW_ * DIM_));
    const int h = ch / 6, c = ch % 6;
    const float* v = qkv + ((size_t)bb * 144 + 96 + h * 6) * HW_ + p;
    const float* wr = attnW + ((size_t)bb * 8 + h) * 36 + c * 6;
    float s = 0.f;
#pragma unroll
    for (int d = 0; d < 6; ++d) s += wr[d] * v[(size_t)d * HW_];
    ao[gid] = s;
}

// ---------------------------------------------------------------------------
// FFN gate, fused: yg = gelu(dw1(y1)) * dw2(y2) * dw3(y3)   (dil 1/2/3, pad=dil)
// ---------------------------------------------------------------------------
__global__ void ffngate_k(const float* __restrict__ y, const float* __restrict__ w1,
                          const float* __restrict__ w2, const float* __restrict__ w3,
                          float* __restrict__ yg)
{
    const size_t gid = (size_t)blockIdx.x * blockDim.x + threadIdx.x;
    if (gid >= (size_t)B_ * 96 * HW_) return;
    const int p = (int)(gid % HW_);
    const int c = (int)((gid / HW_) % 96);
    const int bb = (int)(gid / ((size_t)HW_ * 96));
    const int x = p & (W_ - 1), yv = p >> 8;
    const float* i1 = y + ((size_t)bb * 288 + c) * HW_;
    const float* i2 = i1 + (size_t)96 * HW_;
    const float* i3 = i1 + (size_t)192 * HW_;
    float a1 = 0.f, a2 = 0.f, a3 = 0.f;
#pragma unroll
    for (int ky = 0; ky < 3; ++ky)
#pragma unroll
        for (int kx = 0; kx < 3; ++kx) {
            const int t = ky * 3 + kx;
            int yy = yv + (ky - 1), xx = x + (kx - 1);
            if (yy >= 0 && yy < H_ && xx >= 0 && xx < W_) a1 += w1[c * 9 + t] * i1[yy * W_ + xx];
            yy = yv + (ky - 1) * 2; xx = x + (kx - 1) * 2;
            if (yy >= 0 && yy < H_ && xx >= 0 && xx < W_) a2 += w2[c * 9 + t] * i2[yy * W_ + xx];
            yy = yv + (ky - 1) * 3; xx = x + (kx - 1) * 3;
            if (yy >= 0 && yy < H_ && xx >= 0 && xx < W_) a3 += w3[c * 9 + t] * i3[yy * W_ + xx];
        }
    yg[gid] = gelu_exact(a1) * a2 * a3;
}

// ---------------------------------------------------------------------------
// Conv-descriptor branch stage: conv3x3 s2 p1 + bias -> LN(channels) -> ReLU
// -> maxpool 2x2.  Weights (48*48*9 = 83KB) staged in LDS (320KB/WGP).
// One thread computes one pooled output (4 conv positions sequentially).
// ---------------------------------------------------------------------------
__global__ void conv_stage_k(const float* __restrict__ in, const float* __restrict__ w,
                             const float* __restrict__ bias, const float* __restrict__ lnw,
                             const float* __restrict__ lnb, float* __restrict__ out,
                             int Hi, int Wi)
{
    __shared__ float sw[DIM_ * DIM_ * 9];
    for (int i = threadIdx.x; i < DIM_ * DIM_ * 9; i += blockDim.x) sw[i] = w[i];
    __syncthreads();
    const int Hp = Hi / 4, Wp = Wi / 4;
    const int total = B_ * Hp * Wp;
    const int gid = blockIdx.x * blockDim.x + threadIdx.x;
    if (gid >= total) return;
    const int xp = gid % Wp;
    const int yp = (gid / Wp) % Hp;
    const int bb = gid / (Wp * Hp);
    float pooled[DIM_];
#pragma unroll
    for (int o = 0; o < DIM_; ++o) pooled[o] = -1e30f;
    for (int dy = 0; dy < 2; ++dy)
        for (int dx = 0; dx < 2; ++dx) {
            const int yc = 2 * yp + dy, xc = 2 * xp + dx;
            float acc[DIM_];
#pragma unroll
            for (int o = 0; o < DIM_; ++o) acc[o] = bias[o];
            for (int i = 0; i < DIM_; ++i) {
                float vi[9];
#pragma unroll
                for (int ky = 0; ky < 3; ++ky)
#pragma unroll
                    for (int kx = 0; kx < 3; ++kx) {
                        const int yi = 2 * yc - 1 + ky, xi = 2 * xc - 1 + kx;
                        vi[ky * 3 + kx] = (yi >= 0 && yi < Hi && xi >= 0 && xi < Wi)
                            ? in[(((size_t)bb * DIM_ + i) * Hi + yi) * Wi + xi] : 0.f;
                    }
#pragma unroll 8
                for (int o = 0; o < DIM_; ++o) {
                    const float* wp = &sw[(o * DIM_ + i) * 9];
                    float s = acc[o];
#pragma unroll
                    for (int t = 0; t < 9; ++t) s += wp[t] * vi[t];
                    acc[o] = s;
                }
            }
            float mu = 0.f;
#pragma unroll
            for (int o = 0; o < DIM_; ++o) mu += acc[o];
            mu *= (1.0f / DIM_);
            float var = 0.f;
#pragma unroll
            for (int o = 0; o < DIM_; ++o) { const float d = acc[o] - mu; var += d * d; }
            var *= (1.0f / DIM_);
            const float rstd = rsqrtf(var + 1e-5f);
#pragma unroll
            for (int o = 0; o < DIM_; ++o) {
                float v = (acc[o] - mu) * rstd * lnw[o] + lnb[o];
                v = fmaxf(v, 0.f);
                pooled[o] = fmaxf(pooled[o], v);
            }
        }
#pragma unroll
    for (int o = 0; o < DIM_; ++o)
        out[(((size_t)bb * DIM_ + o) * Hp + yp) * Wp + xp] = pooled[o];
}

// ---------------------------------------------------------------------------
// Descriptor: x_d = gelu(mean(t3 over 4x4)); q0 = x_d @ l2q_w.T + l2q_b
// ---------------------------------------------------------------------------
__global__ void descr_k(const float* __restrict__ t3, const float* __restrict__ l2q_w,
                        const float* __restrict__ l2q_b, float* __restrict__ xd,
                        float* __restrict__ q0)
{
    __shared__ float sxd[96];
    const int tid = threadIdx.x;
    if (tid < 96) {
        float s = 0.f;
        for (int j = 0; j < 16; ++j) s += t3[tid * 16 + j];
        s *= (1.0f / 16.0f);
        const float g = gelu_exact(s);
        sxd[tid] = g;
        xd[tid] = g;
    }
    __syncthreads();
    if (tid < 96) {
        const int bb = tid / 48, c = tid % 48;
        float s = l2q_b[c];
        for (int j = 0; j < 48; ++j) s += sxd[bb * 48 + j] * l2q_w[c * 48 + j];
        q0[tid] = s;
    }
}

// ---------------------------------------------------------------------------
extern "C" void kernel_launch(void* const* d_in, const int* in_sizes, int n_in,
                              void* d_out, int out_size, void* d_ws, size_t ws_size,
                              hipStream_t stream)
{
    (void)in_sizes; (void)n_in; (void)out_size; (void)ws_size;
    const float* x        = (const float*)d_in[0];
    const float* ln1_w    = (const float*)d_in[1];
    const float* ln1_b    = (const float*)d_in[2];
    const float* qkv_w    = (const float*)d_in[3];
    const float* qkv_dw_w = (const float*)d_in[4];
    const float* proj_w   = (const float*)d_in[5];
    const float* fc_w     = (const float*)d_in[6];
    const float* fc_b     = (const float*)d_in[7];
    const float* dep_w    = (const float*)d_in[8];
    const float* dep_b    = (const float*)d_in[9];
    const float* l2q_w    = (const float*)d_in[10];
    const float* l2q_b    = (const float*)d_in[11];
    const float* temp     = (const float*)d_in[12];
    const float* ln2_w    = (const float*)d_in[13];
    const float* ln2_b    = (const float*)d_in[14];
    const float* pin_w    = (const float*)d_in[15];
    const float* dw1_w    = (const float*)d_in[16];
    const float* dw2_w    = (const float*)d_in[17];
    const float* dw3_w    = (const float*)d_in[18];
    const float* pout_w   = (const float*)d_in[19];
    const float* cv_w[3]  = {(const float*)d_in[20], (const float*)d_in[24], (const float*)d_in[28]};
    const float* cv_b[3]  = {(const float*)d_in[21], (const float*)d_in[25], (const float*)d_in[29]};
    const float* cln_w[3] = {(const float*)d_in[22], (const float*)d_in[26], (const float*)d_in[30]};
    const float* cln_b[3] = {(const float*)d_in[23], (const float*)d_in[27], (const float*)d_in[31]};

    float* ws  = (float*)d_ws;
    float* X2  = ws + OFF_X2;
    float* XN  = ws + OFF_XN;
    float* OC  = ws + OFF_OC;
    float* AO  = ws + OFF_AO;
    float* QB  = ws + OFF_Q;     // qkv post-dw; later yg
    float* BIG = ws + OFF_BIG;   // qkv_pre -> f_conv -> y
    float* T1  = ws + OFF_T1;
    float* T2  = ws + OFF_T2;
    float* T3  = ws + OFF_T3;
    float* XD  = ws + OFF_XD;
    float* Q0  = ws + OFF_Q0;
    float* NRM = ws + OFF_NRM;
    float* AW  = ws + OFF_AW;
    float* AP  = ws + OFF_AP;
    float* OUT = (float*)d_out;

    auto cdiv = [](size_t a, size_t b) { return (unsigned)((a + b - 1) / b); };

    // ---- conv descriptor branch
    conv_stage_k<<<cdiv((size_t)B_ * 64 * 64, 128), 128, 0, stream>>>(
        x, cv_w[0], cv_b[0], cln_w[0], cln_b[0], T1, 256, 256);
    conv_stage_k<<<cdiv((size_t)B_ * 16 * 16, 128), 128, 0, stream>>>(
        T1, cv_w[1], cv_b[1], cln_w[1], cln_b[1], T2, 64, 64);
    conv_stage_k<<<1, 128, 0, stream>>>(
        T2, cv_w[2], cv_b[2], cln_w[2], cln_b[2], T3, 16, 16);
    descr_k<<<1, 96, 0, stream>>>(T3, l2q_w, l2q_b, XD, Q0);

    // ---- attention
    ln_k<<<cdiv((size_t)B_ * HW_, 256), 256, 0, stream>>>(x, ln1_w, ln1_b, XN);
    wmma_mm_k<<<dim3(HW_ / 64, 144 / 16, B_), 128, 0, stream>>>(
        qkv_w, XN, BIG, nullptr, nullptr, nullptr, 144, 48);                 // qkv 1x1
    dwconv3_k<<<cdiv((size_t)B_ * 144 * HW_, 256), 256, 0, stream>>>(
        BIG, qkv_dw_w, QB, 144, 1);                                          // qkv depthwise
    fc_k<<<dim3(HW_ / 64, B_), 64, 0, stream>>>(QB, fc_w, fc_b, BIG);        // f_conv
    depconv_k<<<cdiv((size_t)B_ * DIM_ * HW_, 256), 256, 0, stream>>>(
        BIG, dep_w, dep_b, OC);                                              // out_conv
    rownorm_k<<<B_ * 96, 256, 0, stream>>>(QB, NRM);
    gram_k<<<dim3(16, GSPLIT), 256, 0, stream>>>(QB, AP);
    softmax_k<<<1, 96, 0, stream>>>(AP, NRM, temp, Q0, AW);
    attnv_k<<<cdiv((size_t)B_ * DIM_ * HW_, 256), 256, 0, stream>>>(QB, AW, AO);
    wmma_mm_k<<<dim3(HW_ / 64, DIM_ / 16, B_), 128, 0, stream>>>(
        proj_w, AO, X2, nullptr, x, OC, DIM_, DIM_);                         // x + proj + out_conv

    // ---- feed-forward
    ln_k<<<cdiv((size_t)B_ * HW_, 256), 256, 0, stream>>>(X2, ln2_w, ln2_b, XN);
    wmma_mm_k<<<dim3(HW_ / 64, 288 / 16, B_), 128, 0, stream>>>(
        pin_w, XN, BIG, nullptr, nullptr, nullptr, 288, 48);                 // pin
    ffngate_k<<<cdiv((size_t)B_ * 96 * HW_, 256), 256, 0, stream>>>(
        BIG, dw1_w, dw2_w, dw3_w, QB);                                       // gated yg
    wmma_mm_k<<<dim3(HW_ / 64, DIM_ / 16, B_), 128, 0, stream>>>(
        pout_w, QB, OUT, nullptr, X2, nullptr, DIM_, 96);                    // X2 + pout(yg)
}